// Router_30923764531755
// MI455X (gfx1250) — compile-verified
//
#include <hip/hip_runtime.h>
#include <cstddef>

// Problem constants (from reference): T=8192 tokens, D=1024, E=8 experts, CAP=2048
#define TT        8192
#define DD        1024
#define EE        8
#define CAPACITY  2048
#define TOK_PER_BLOCK 64         // tokens per router block (4 waves x 16-token WMMA tiles)
#define NB        (TT / TOK_PER_BLOCK)   // 128 blocks

typedef __attribute__((ext_vector_type(2))) float v2f;
typedef __attribute__((ext_vector_type(4))) float f4;
typedef __attribute__((ext_vector_type(8))) float v8f;

// ---------------------------------------------------------------------------
// Kernel A: router logits via V_WMMA_F32_16X16X4_F32, softmax over E=8,
// top-1 gate/index, per-block expert histogram + prob sums (fixed-order,
// deterministic reductions — no float atomics).
// Block = 128 threads = 4 waves; each wave owns a 16-token x 16-expert tile.
// B columns 8..15 simply replicate columns 0..7 (their C columns are never
// read back), so every load in the hot loop is unconditional — no EXEC
// divergence around the WMMAs.
// ---------------------------------------------------------------------------
__global__ void __launch_bounds__(128)
router_logits_kernel(const float* __restrict__ x, const float* __restrict__ W,
                     const float* __restrict__ bias,
                     float* __restrict__ gate, int* __restrict__ eidx,
                     int* __restrict__ blockHist, float* __restrict__ blockProbSum)
{
    __shared__ float lds_logit[4][16][EE];        // [wave][m][e]
    __shared__ float lds_p[TOK_PER_BLOCK][EE];    // per-token probs
    __shared__ int   lds_e[TOK_PER_BLOCK];        // per-token top-1 expert

    const int tid  = threadIdx.x;
    const int wave = tid >> 5;
    const int lane = tid & 31;
    const int mrow = lane & 15;     // token row within 16-tile
    const int half = lane >> 4;     // A/B K-half select (K+0,1 vs K+2,3)
    const int ncol = lane & 15;     // expert column for B / C
    const int wcol = ncol & 7;      // replicated W column (cols 8..15 are dummies)

    const int tile = blockIdx.x * 4 + wave;                 // 16-token tile id
    const float* xr = x + (size_t)(tile * 16 + mrow) * DD + 2 * half;
    const float* wr = W + (size_t)(2 * half) * EE + wcol;   // row (4k + 2*half), col wcol

    v8f c0 = {0.f,0.f,0.f,0.f,0.f,0.f,0.f,0.f};
    v8f c1 = {0.f,0.f,0.f,0.f,0.f,0.f,0.f,0.f};

    // K loop: 1024 / 4 = 256 WMMA steps, 2 independent accumulator chains.
    for (int k = 0; k < DD / 4; k += 2) {
        v2f a0, a1, b0, b1;
        // A fragment: lane<16 -> K = 4k+0,1 ; lane>=16 -> K = 4k+2,3 (ISA layout)
        a0[0] = xr[4*k + 0]; a0[1] = xr[4*k + 1];
        a1[0] = xr[4*k + 4]; a1[1] = xr[4*k + 5];
        // B fragment: W rows kk, kk+1 (stride E floats), unconditional loads
        b0[0] = wr[(4*k)     * EE]; b0[1] = wr[(4*k + 1) * EE];
        b1[0] = wr[(4*k + 4) * EE]; b1[1] = wr[(4*k + 5) * EE];
        c0 = __builtin_amdgcn_wmma_f32_16x16x4_f32(false, a0, false, b0,
                                                   (short)0, c0, false, false);
        c1 = __builtin_amdgcn_wmma_f32_16x16x4_f32(false, a1, false, b1,
                                                   (short)0, c1, false, false);
    }
    v8f c = c0 + c1;

    // C layout: lane holds C[m = r + 8*half][n = ncol] in c[r]; cols >= 8 are
    // replicas of cols 0..7 and are simply not stored.
    if (ncol < EE) {
        #pragma unroll
        for (int r = 0; r < 8; ++r)
            lds_logit[wave][r + 8 * half][ncol] = c[r];
    }
    __syncthreads();

    // Softmax + top-1 (one thread per token; first-index tie-break like top_k)
    if (tid < TOK_PER_BLOCK) {
        const int w = tid >> 4, m = tid & 15;
        float l[EE], p[EE];
        float mx = -3.4e38f;
        #pragma unroll
        for (int e = 0; e < EE; ++e) { l[e] = lds_logit[w][m][e] + bias[e]; mx = fmaxf(mx, l[e]); }
        float s = 0.f;
        #pragma unroll
        for (int e = 0; e < EE; ++e) { p[e] = expf(l[e] - mx); s += p[e]; }
        const float inv = 1.f / s;
        int best = 0; float bp = -1.f;
        #pragma unroll
        for (int e = 0; e < EE; ++e) {
            p[e] *= inv;
            lds_p[tid][e] = p[e];
            if (p[e] > bp) { bp = p[e]; best = e; }
        }
        lds_e[tid] = best;
        const int t = blockIdx.x * TOK_PER_BLOCK + tid;
        gate[t] = bp;
        eidx[t] = best;
    }
    __syncthreads();

    // Deterministic per-block reductions: one thread per expert, fixed order.
    if (tid < EE) {
        int cnt = 0; float ps = 0.f;
        for (int j = 0; j < TOK_PER_BLOCK; ++j) {
            cnt += (lds_e[j] == tid) ? 1 : 0;
            ps  += lds_p[j][tid];
        }
        blockHist[blockIdx.x * EE + tid]    = cnt;
        blockProbSum[blockIdx.x * EE + tid] = ps;
    }
}

// ---------------------------------------------------------------------------
// Kernel B: zero the 1 GB dispatch+combined output with streaming b128 stores.
// Output >> L2 (192 MB), so non-temporal is the right hint. This is the
// roofline term: ~1 GB / 23.3 TB/s ~= 43 us.
// ---------------------------------------------------------------------------
__global__ void zero_out_kernel(f4* __restrict__ out, size_t n4)
{
    size_t i = (size_t)blockIdx.x * blockDim.x + threadIdx.x;
    const size_t stride = (size_t)gridDim.x * blockDim.x;
    f4 z = {0.f, 0.f, 0.f, 0.f};
    for (; i < n4; i += stride)
        __builtin_nontemporal_store(z, out + i);
}

// ---------------------------------------------------------------------------
// Kernel C: exclusive prefix over block histograms (per expert), totals,
// prob sums, and the aux load-balance loss (fixed order -> deterministic).
// aux = E^2 * mean_e((sumP_e/T) * (cnt_e/T)) = 8 * sum_e(...)
// ---------------------------------------------------------------------------
__global__ void prefix_aux_kernel(const int* __restrict__ blockHist,
                                  const float* __restrict__ blockProbSum,
                                  int* __restrict__ blockBase,
                                  float* __restrict__ aux_out)
{
    __shared__ int   tot[EE];
    __shared__ float ps[EE];
    const int e = threadIdx.x;
    if (e < EE) {
        int run = 0; float s = 0.f;
        for (int bl = 0; bl < NB; ++bl) {
            blockBase[bl * EE + e] = run;
            run += blockHist[bl * EE + e];
            s   += blockProbSum[bl * EE + e];
        }
        tot[e] = run; ps[e] = s;
    }
    __syncthreads();
    if (threadIdx.x == 0) {
        const float invT = 1.0f / (float)TT;
        float aux = 0.f;
        for (int k = 0; k < EE; ++k)
            aux += (ps[k] * invT) * ((float)tot[k] * invT);
        aux *= (float)(EE * EE) / (float)EE;   // mean over E, then * E^2
        *aux_out = aux;
    }
}

// ---------------------------------------------------------------------------
// Kernel D: scatter. pos1 = 1-indexed cumulative count per expert
//   = blockBase + within-block rank + 1; kept iff pos1 < CAP;
// one_hot index c = pos1 (matches reference's 1-indexed one_hot, where slot 0
// is never used). Writes dispatch and combined (identical tensors).
// ---------------------------------------------------------------------------
__global__ void __launch_bounds__(TOK_PER_BLOCK)
scatter_kernel(const float* __restrict__ gate, const int* __restrict__ eidx,
               const int* __restrict__ blockBase, float* __restrict__ out)
{
    __shared__ int lidx[TOK_PER_BLOCK];
    const int tid = threadIdx.x;
    const int t   = blockIdx.x * TOK_PER_BLOCK + tid;
    const int e   = eidx[t];
    lidx[tid] = e;
    __syncthreads();
    int rank = 0;
    for (int j = 0; j < tid; ++j) rank += (lidx[j] == e) ? 1 : 0;
    const int pos1 = blockBase[blockIdx.x * EE + e] + rank + 1;
    if (pos1 < CAPACITY) {
        const float v = gate[t];
        const size_t off = ((size_t)t * EE + e) * CAPACITY + (size_t)pos1;
        out[off] = v;
        out[(size_t)TT * EE * CAPACITY + off] = v;
    }
}

// ---------------------------------------------------------------------------
extern "C" void kernel_launch(void* const* d_in, const int* in_sizes, int n_in,
                              void* d_out, int out_size, void* d_ws, size_t ws_size,
                              hipStream_t stream)
{
    (void)in_sizes; (void)n_in; (void)out_size; (void)ws_size;

    const float* x = (const float*)d_in[0];   // [T, D] f32
    const float* W = (const float*)d_in[1];   // [D, E] f32
    const float* b = (const float*)d_in[2];   // [E]    f32
    // d_in[3] = expert_capacity scalar (known constant 2048)

    float* out = (float*)d_out;               // dispatch | combined | aux

    // Workspace layout (~76 KB)
    char*  ws           = (char*)d_ws;
    float* gate         = (float*)(ws);                             // T f32
    int*   eidx         = (int*)  (ws + (size_t)TT * 4);            // T i32
    int*   blockHist    = (int*)  (ws + (size_t)TT * 8);            // NB*E
    float* blockProbSum = (float*)(ws + (size_t)TT * 8 + NB*EE*4);  // NB*E
    int*   blockBase    = (int*)  (ws + (size_t)TT * 8 + 2*NB*EE*4);// NB*E

    const size_t TEC = (size_t)TT * EE * CAPACITY;   // 134,217,728 elements

    // 1) WMMA logits + softmax + top-1 + per-block stats (reads 32 MB of x)
    router_logits_kernel<<<NB, 128, 0, stream>>>(x, W, b, gate, eidx,
                                                 blockHist, blockProbSum);
    // 2) Zero-fill the 1 GB dispatch+combined output (the roofline term)
    zero_out_kernel<<<4096, 256, 0, stream>>>((f4*)out, (2 * TEC) / 4);
    // 3) Cross-block exclusive prefix + aux loss (scalar at out[2*TEC])
    prefix_aux_kernel<<<1, 32, 0, stream>>>(blockHist, blockProbSum, blockBase,
                                            out + 2 * TEC);
    // 4) Scatter 8192 gate values into each of the two tensors
    scatter_kernel<<<NB, TOK_PER_BLOCK, 0, stream>>>(gate, eidx, blockBase, out);
}